// LocationProbabilityMatchingLoss_89575837925967
// MI455X (gfx1250) — compile-verified
//
#include <hip/hip_runtime.h>
#include <hip/hip_bf16.h>
#include <math.h>

typedef __attribute__((ext_vector_type(2))) float v2f;
typedef __attribute__((ext_vector_type(8))) float v8f;

#define BB 8
#define HH 64
#define WW 64
#define NN 128
#define HW 4096
#define KSPLIT 8
#define KCHUNK (HW / KSPLIT)   // 512

// ---------------------------------------------------------------------------
// Kernel 1: partial 16x16 Gram matrices via V_WMMA_F32_16X16X4_F32.
// Work id = gg*KSPLIT + s : gg = dir*256 + b*32 + g (512 groups), s = K-chunk.
// 4096 waves total (512 blocks x 8 waves) -> enough outstanding loads to
// approach the HBM bandwidth-delay product; K-chunks are disjoint so partials
// simply add. A-operand (ISA 7.12.2, 32-bit A 16x4): lane m / m+16 hold row m;
// VGPR0 = K+0 / K+2, VGPR1 = K+1 / K+3. For Gram A*A^T the B operand register
// image is identical, so the same v2f fragment is passed twice.
// ---------------------------------------------------------------------------
__global__ __launch_bounds__(256) void gram_kernel(const float* __restrict__ corr,
                                                   const float* __restrict__ kp,
                                                   float* __restrict__ gram_part) {
  const int wave = threadIdx.x >> 5;
  const int lane = threadIdx.x & 31;
  const int wid  = blockIdx.x * 8 + wave;   // 0..4095
  const int s    = wid & (KSPLIT - 1);
  const int gg   = wid >> 3;                // 0..511
  const int dir  = gg >> 8;
  const int rest = gg & 255;
  const int b    = rest >> 5;
  const int g    = rest & 31;
  const int m    = lane & 15;
  const int n    = g * 4 + (m >> 2);
  const int k    = m & 3;

  // prediction keypoint for this direction: kp[b,n,(y|x),dir]
  const float py = kp[((b * NN + n) * 2 + 0) * 2 + dir];
  const float px = kp[((b * NN + n) * 2 + 1) * 2 + dir];
  int y0 = (int)floorf(py); y0 = min(max(y0, 0), HH - 1);
  int x0 = (int)floorf(px); x0 = min(max(x0, 0), WW - 1);
  const int y1 = min(y0 + 1, HH - 1);
  const int x1 = min(x0 + 1, WW - 1);
  const int cy = (k < 2) ? y0 : y1;     // corners: (y0,x0),(y0,x1),(y1,x0),(y1,x1)
  const int cx = (k & 1) ? x1 : x0;
  const int koff = (lane >= 16) ? 2 : 0;
  const int kbeg = s * KCHUNK;
  const int kend = kbeg + KCHUNK;

  v8f acc = {};
  if (dir == 0) {
    // row = corr[b, cy, cx, :] : contiguous -> b64/lane/step, non-temporal
    // (single use; keep L2 capacity for the dir-1 gather working set)
    const v2f* p2 = (const v2f*)(corr + (((size_t)((b * HH + cy) * WW + cx)) << 12));
#pragma unroll 8
    for (int kb = kbeg; kb < kend; kb += 4) {
      v2f a = __builtin_nontemporal_load(&p2[(kb + koff) >> 1]);
      acc = __builtin_amdgcn_wmma_f32_16x16x4_f32(false, a, false, a,
                                                  (short)0, acc, false, false);
    }
  } else {
    // column = corr[b, :, :, cy, cx] : 16 KB stride -> L2-resident gather
    const float* p = corr + (((size_t)b) << 24) + (size_t)(cy * WW + cx);
#pragma unroll 8
    for (int kb = kbeg; kb < kend; kb += 4) {
      const int c = kb + koff;
      v2f a;
      a.x = p[(size_t)c << 12];
      a.y = p[(size_t)(c + 1) << 12];
      acc = __builtin_amdgcn_wmma_f32_16x16x4_f32(false, a, false, a,
                                                  (short)0, acc, false, false);
    }
  }

  // C/D layout: VGPR v, lane l -> row M = v + (l<16?0:8), col N = l&15
  float* gw = gram_part + (size_t)wid * 256;
  const int col   = lane & 15;
  const int rbase = (lane < 16) ? 0 : 8;
#pragma unroll
  for (int v = 0; v < 8; ++v) gw[(rbase + v) * 16 + col] = acc[v];
}

// ---------------------------------------------------------------------------
// Kernel 2: per-(dir,b,n) inner term:  w^T G w  - 2*<pred,target> + ||target||^2
// G is summed over the KSPLIT partials while contracting the 4x4 block.
// target support is a 6x6 window (bilinear spike conv 5x5 Gaussian).
// ---------------------------------------------------------------------------
__global__ __launch_bounds__(256) void item_kernel(const float* __restrict__ corr,
                                                   const float* __restrict__ kp,
                                                   const float* __restrict__ gram_part,
                                                   float* __restrict__ inner) {
  const int item = blockIdx.x * blockDim.x + threadIdx.x;  // 0..2047
  if (item >= 2048) return;
  const int dir = item >> 10;
  const int b   = (item >> 7) & 7;
  const int n   = item & 127;

  const float py = kp[((b * NN + n) * 2 + 0) * 2 + dir];        // pred pts
  const float px = kp[((b * NN + n) * 2 + 1) * 2 + dir];
  const float ty = kp[((b * NN + n) * 2 + 0) * 2 + (1 - dir)];  // target pts
  const float tx = kp[((b * NN + n) * 2 + 1) * 2 + (1 - dir)];

  // ---- prediction bilinear corners & weights (matches _interp_rows) ----
  const float fyp = floorf(py), fxp = floorf(px);
  const float wy = py - fyp, wx = px - fxp;
  int y0 = (int)fyp; y0 = min(max(y0, 0), HH - 1);
  int x0 = (int)fxp; x0 = min(max(x0, 0), WW - 1);
  const int y1 = min(y0 + 1, HH - 1);
  const int x1 = min(x0 + 1, WW - 1);
  float w[4]  = {(1.f - wy) * (1.f - wx), (1.f - wy) * wx, wy * (1.f - wx), wy * wx};
  int pcy[4] = {y0, y0, y1, y1};
  int pcx[4] = {x0, x1, x0, x1};

  // ---- ||pred||^2 = w^T G w ; G = sum of KSPLIT partial Grams ----
  const int gg = dir * 256 + b * 32 + (n >> 2);
  const int m0 = (n & 3) * 4;
  float predsq = 0.f;
  for (int s = 0; s < KSPLIT; ++s) {
    const float* G = gram_part + ((size_t)(gg * KSPLIT + s)) * 256;
#pragma unroll
    for (int a = 0; a < 4; ++a)
#pragma unroll
      for (int c = 0; c < 4; ++c)
        predsq += w[a] * w[c] * G[(m0 + a) * 16 + (m0 + c)];
  }

  // ---- 5-tap Gaussian (size 5, sigma = 2/3) ----
  float kv[5]; float ksum = 0.f;
#pragma unroll
  for (int i = 0; i < 5; ++i) {
    kv[i] = expf(-(float)((i - 2) * (i - 2)) * (9.0f / 8.0f));
    ksum += kv[i];
  }
#pragma unroll
  for (int i = 0; i < 5; ++i) kv[i] /= ksum;

  // ---- target spikes (floor/ceil, distance-proportional weights) ----
  const float tfy = floorf(ty), tcy = ceilf(ty);
  const float tfx = floorf(tx), tcx = ceilf(tx);
  float nyv[4] = {tfy, tfy, tcy, tcy};
  float nxv[4] = {tfx, tcx, tfx, tcx};
  float d[4]; float dsum = 0.f;
#pragma unroll
  for (int a = 0; a < 4; ++a) {
    const float dy = nyv[a] - ty + 1e-6f;
    const float dx = nxv[a] - tx + 1e-6f;
    d[a] = sqrtf(dy * dy + dx * dx);
    dsum += d[a];
  }
  const float dnorm = fmaxf(dsum, 1e-12f);

  // ---- 6x6 window = spikes convolved with 5x5 separable Gaussian ----
  float win[6][6];
#pragma unroll
  for (int iy = 0; iy < 6; ++iy)
#pragma unroll
    for (int ix = 0; ix < 6; ++ix) win[iy][ix] = 0.f;
  const int ay = (int)tfy - 2, ax = (int)tfx - 2;
#pragma unroll
  for (int a = 0; a < 4; ++a) {
    const float t = d[a] / dnorm;
    const int sy = (int)nyv[a] - ay - 2;   // 0 or 1
    const int sx = (int)nxv[a] - ax - 2;
#pragma unroll
    for (int dy = 0; dy < 5; ++dy)
#pragma unroll
      for (int dx = 0; dx < 5; ++dx)
        win[sy + dy][sx + dx] += t * kv[dy] * kv[dx];
  }

  // ---- cross term and target norm over in-bounds support cells ----
  size_t basek[4];
  size_t cstride;
  if (dir == 0) {
    cstride = 1;
#pragma unroll
    for (int a = 0; a < 4; ++a)
      basek[a] = ((size_t)((b * HH + pcy[a]) * WW + pcx[a])) << 12;
  } else {
    cstride = 4096;
#pragma unroll
    for (int a = 0; a < 4; ++a)
      basek[a] = (((size_t)b) << 24) + (size_t)(pcy[a] * WW + pcx[a]);
  }

  float s2 = 0.f, cross = 0.f;
  for (int iy = 0; iy < 6; ++iy) {
    const int gy = ay + iy;
    if (gy < 0 || gy >= HH) continue;
    for (int ix = 0; ix < 6; ++ix) {
      const int gx = ax + ix;
      if (gx < 0 || gx >= WW) continue;
      const float v = win[iy][ix];
      s2 += v * v;
      const size_t cell = (size_t)(gy * WW + gx) * cstride;
      float pr = 0.f;
#pragma unroll
      for (int a = 0; a < 4; ++a) pr += w[a] * corr[basek[a] + cell];
      cross += v * pr;
    }
  }
  const float nrm = fmaxf(sqrtf(s2), 1e-12f);
  const float tsq = s2 / (nrm * nrm);   // == 1 when norm above eps
  inner[item] = predsq - 2.f * (cross / nrm) + tsq;
}

// ---------------------------------------------------------------------------
// Kernel 3: deterministic reduce: loss = (1/B) * sum_{dir,b} sqrt(sum_n inner)
// ---------------------------------------------------------------------------
__global__ void reduce_kernel(const float* __restrict__ inner, float* __restrict__ out) {
  __shared__ float sh[16];
  const int t = threadIdx.x;
  if (t < 16) {
    float s = 0.f;
    for (int i = 0; i < 128; ++i) s += inner[t * 128 + i];
    sh[t] = sqrtf(fmaxf(s, 0.f));
  }
  __syncthreads();
  if (t == 0) {
    float tot = 0.f;
    for (int i = 0; i < 16; ++i) tot += sh[i];
    out[0] = tot / (float)BB;
  }
}

extern "C" void kernel_launch(void* const* d_in, const int* in_sizes, int n_in,
                              void* d_out, int out_size, void* d_ws, size_t ws_size,
                              hipStream_t stream) {
  const float* corr = (const float*)d_in[0];
  const float* kp   = (const float*)d_in[1];
  float* gram_part = (float*)d_ws;                      // 512*8*256 floats = 4 MB
  float* inner     = gram_part + 512 * KSPLIT * 256;    // 2048 floats
  float* out       = (float*)d_out;

  gram_kernel<<<512, 256, 0, stream>>>(corr, kp, gram_part);
  item_kernel<<<8, 256, 0, stream>>>(corr, kp, gram_part, inner);
  reduce_kernel<<<1, 32, 0, stream>>>(inner, out);
}